// HardMiningLoss_26860725469565
// MI455X (gfx1250) — compile-verified
//
#include <hip/hip_runtime.h>
#include <hip/hip_bf16.h>

// HardMiningLoss, algebraically collapsed:
//   loss = mean_i [ 1 - s_i + (u_i - cnt_i*s_i)/(N - cnt_i) ]
//   s_i = <x_i, w[t_i]>,  u_i = <x_i, W_sum>,  W_sum = sum_c cnt[c]*w[c]
// s_i/u_i computed 16 rows at a time with v_wmma_f32_16x16x4_f32 (fp32-exact).

typedef float v2f __attribute__((ext_vector_type(2)));
typedef float v8f __attribute__((ext_vector_type(8)));

#define TILE_M 16
#define WAVES_PER_TILE 8
#define NCHUNK 16

// ---------------- Kernel 0: zero class counters ----------------
__global__ void k_zero_cnt(int* __restrict__ cnt, int C) {
  int c = blockIdx.x * blockDim.x + threadIdx.x;
  if (c < C) cnt[c] = 0;
}

// ---------------- Kernel 1: class histogram of targets ----------------
__global__ void k_hist(const int* __restrict__ tgt, int* __restrict__ cnt, int N) {
  int i = blockIdx.x * blockDim.x + threadIdx.x;
  if (i < N) atomicAdd(&cnt[tgt[i]], 1);
}

// ---------------- Kernel 2a: W_sum partials over class chunks ----------------
// wpart[chunk*D + d] = sum_{c in chunk} cnt[c] * w[c*D + d]   (skip cnt==0 rows)
__global__ void k_wsum_part(const float* __restrict__ w, const int* __restrict__ cnt,
                            float* __restrict__ wpart, int C, int D, int cpc) {
  int d = blockIdx.x * blockDim.x + threadIdx.x;
  int chunk = blockIdx.y;
  if (d >= D) return;
  int c0 = chunk * cpc;
  int c1 = c0 + cpc; if (c1 > C) c1 = C;
  float acc = 0.f;
  for (int c = c0; c < c1; ++c) {
    int n = cnt[c];
    if (n != 0) acc += (float)n * w[(size_t)c * D + d];
  }
  wpart[(size_t)chunk * D + d] = acc;
}

// ---------------- Kernel 2b: reduce partials -> W_sum ----------------
__global__ void k_wsum_reduce(const float* __restrict__ wpart, float* __restrict__ wsum,
                              int D, int nchunks) {
  int d = blockIdx.x * blockDim.x + threadIdx.x;
  if (d >= D) return;
  float acc = 0.f;
  for (int q = 0; q < nchunks; ++q) acc += wpart[(size_t)q * D + d];
  wsum[d] = acc;
}

// ---------------- Kernel 3: per-16-row-tile loss via WMMA ----------------
// Block = WAVES_PER_TILE waves, one tile of 16 rows. Each wave handles a K-slice
// of D, accumulating acc1 = X_tile*G_tile^T and acc2 = X_tile*[W_sum broadcast].
// Diagonals give s_i and u_i; combine with counts; one partial per tile.
__global__ void __launch_bounds__(32 * WAVES_PER_TILE)
k_loss_tiles(const float* __restrict__ X, const int* __restrict__ tgt,
             const float* __restrict__ W, const int* __restrict__ cnt,
             const float* __restrict__ wsum, float* __restrict__ tilepart,
             int N, int D) {
  __shared__ float s_s[WAVES_PER_TILE][TILE_M];
  __shared__ float s_u[WAVES_PER_TILE][TILE_M];
  __shared__ float s_loss[TILE_M];

  const int lane = threadIdx.x & 31;
  const int wid  = threadIdx.x >> 5;
  const int tile = blockIdx.x;
  const int row0 = tile * TILE_M;

  // f32 WMMA 16x16x4 layouts: A(16x4): lane holds M=lane%16, k = vgpr + 2*(lane>=16).
  // B(4x16): lane holds N=lane%16, same k mapping. So per lane both A and B need a
  // contiguous float2 from (row = row0 + lane%16) at column k + 2*(lane>=16).
  const int rloc  = lane & 15;
  const int khalf = (lane >> 4) << 1;  // 0 or 2
  const int row   = row0 + rloc;

  const int kspan = D / WAVES_PER_TILE;   // per-wave K slice (1024/8 = 128)
  const int koff  = wid * kspan + khalf;

  // Fold the wave's K offset into the base pointers ONCE so the loop trip count
  // below is wave-uniform (scalar loop, no exec masking, unrollable; also keeps
  // EXEC == all-ones around every WMMA as the ISA requires).
  const float* xp = X    + (size_t)row * D      + koff;
  const float* gp = W    + (size_t)tgt[row] * D + koff;  // gathered proxy row
  const float* wp = wsum + koff;

  v8f acc1 = {};  // X * G^T  -> diag = s_i (partial over this wave's K slice)
  v8f acc2 = {};  // X * Wsum -> diag = u_i (every column equals u_i)

  const int ksteps = kspan >> 2;          // K=4 per WMMA; uniform trip count
  int it = 0;
  // Manual 4x unroll: issue all 12 b64 loads, then 8 back-to-back WMMAs so the
  // loads of the next group overlap the matrix pipe.
  for (; it + 4 <= ksteps; it += 4) {
    const float* x0 = xp;  const float* g0 = gp;  const float* w0 = wp;
    v2f a0 = *(const v2f*)(x0 + 0);   v2f b0 = *(const v2f*)(g0 + 0);   v2f c0 = *(const v2f*)(w0 + 0);
    v2f a1 = *(const v2f*)(x0 + 4);   v2f b1 = *(const v2f*)(g0 + 4);   v2f c1 = *(const v2f*)(w0 + 4);
    v2f a2 = *(const v2f*)(x0 + 8);   v2f b2 = *(const v2f*)(g0 + 8);   v2f c2 = *(const v2f*)(w0 + 8);
    v2f a3 = *(const v2f*)(x0 + 12);  v2f b3 = *(const v2f*)(g0 + 12);  v2f c3 = *(const v2f*)(w0 + 12);
    xp += 16; gp += 16; wp += 16;
    acc1 = __builtin_amdgcn_wmma_f32_16x16x4_f32(false, a0, false, b0, (short)0, acc1, false, false);
    acc2 = __builtin_amdgcn_wmma_f32_16x16x4_f32(false, a0, false, c0, (short)0, acc2, false, false);
    acc1 = __builtin_amdgcn_wmma_f32_16x16x4_f32(false, a1, false, b1, (short)0, acc1, false, false);
    acc2 = __builtin_amdgcn_wmma_f32_16x16x4_f32(false, a1, false, c1, (short)0, acc2, false, false);
    acc1 = __builtin_amdgcn_wmma_f32_16x16x4_f32(false, a2, false, b2, (short)0, acc1, false, false);
    acc2 = __builtin_amdgcn_wmma_f32_16x16x4_f32(false, a2, false, c2, (short)0, acc2, false, false);
    acc1 = __builtin_amdgcn_wmma_f32_16x16x4_f32(false, a3, false, b3, (short)0, acc1, false, false);
    acc2 = __builtin_amdgcn_wmma_f32_16x16x4_f32(false, a3, false, c3, (short)0, acc2, false, false);
  }
  for (; it < ksteps; ++it) {           // remainder (not taken for D=1024)
    v2f a = *(const v2f*)(xp);  v2f b = *(const v2f*)(gp);  v2f c = *(const v2f*)(wp);
    xp += 4; gp += 4; wp += 4;
    acc1 = __builtin_amdgcn_wmma_f32_16x16x4_f32(false, a, false, b, (short)0, acc1, false, false);
    acc2 = __builtin_amdgcn_wmma_f32_16x16x4_f32(false, a, false, c, (short)0, acc2, false, false);
  }

  // Diagonal extraction from documented 16x16 f32 C/D layout:
  //   m < 8 : lane m,     vgpr m
  //   m >= 8: lane m+16,  vgpr m-8
  int m = -1, vsel = 0;
  if (lane < 8)        { m = lane;      vsel = lane;      }
  else if (lane >= 24) { m = lane - 16; vsel = lane - 24; }
  float s = 0.f, u = 0.f;
#pragma unroll
  for (int v = 0; v < 8; ++v) {
    if (v == vsel) { s = acc1[v]; u = acc2[v]; }
  }
  if (m >= 0) { s_s[wid][m] = s; s_u[wid][m] = u; }
  __syncthreads();

  if (threadIdx.x < TILE_M) {
    const int mm = threadIdx.x;
    float S = 0.f, U = 0.f;
#pragma unroll
    for (int q = 0; q < WAVES_PER_TILE; ++q) { S += s_s[q][mm]; U += s_u[q][mm]; }
    const int p  = cnt[tgt[row0 + mm]];      // positives incl. diagonal (>= 1)
    const float pn = (float)(N - p);         // negatives per row
    s_loss[mm] = 1.f - S + (U - (float)p * S) / pn;
  }
  __syncthreads();

  if (threadIdx.x == 0) {
    float t = 0.f;
    for (int mm = 0; mm < TILE_M; ++mm) t += s_loss[mm];
    tilepart[tile] = t;
  }
}

// ---------------- Kernel 4: deterministic final reduction ----------------
__global__ void k_final(const float* __restrict__ tilepart, int ntiles, float inv_n,
                        float* __restrict__ out) {
  __shared__ float sh[256];
  float v = 0.f;
  for (int i = threadIdx.x; i < ntiles; i += 256) v += tilepart[i];
  sh[threadIdx.x] = v;
  __syncthreads();
  for (int s = 128; s > 0; s >>= 1) {
    if ((int)threadIdx.x < s) sh[threadIdx.x] += sh[threadIdx.x + s];
    __syncthreads();
  }
  if (threadIdx.x == 0) out[0] = sh[0] * inv_n;
}

extern "C" void kernel_launch(void* const* d_in, const int* in_sizes, int n_in,
                              void* d_out, int out_size, void* d_ws, size_t ws_size,
                              hipStream_t stream) {
  const float* X = (const float*)d_in[0];   // [N, D] fp32
  const int*   T = (const int*)d_in[1];     // [N] int (harness convention for integer dtype)
  const float* W = (const float*)d_in[2];   // [C, D] fp32
  float* out = (float*)d_out;

  const int N = in_sizes[1];
  const int D = in_sizes[0] / N;            // 1024
  const int C = in_sizes[2] / D;            // 10000

  // Workspace layout (256B-aligned sections)
  char* ws = (char*)d_ws;
  size_t off = 0;
  int* cnt = (int*)(ws + off);
  off = (off + (size_t)C * sizeof(int) + 255) & ~(size_t)255;
  float* wpart = (float*)(ws + off);
  off = (off + (size_t)NCHUNK * D * sizeof(float) + 255) & ~(size_t)255;
  float* wsum = (float*)(ws + off);
  off = (off + (size_t)D * sizeof(float) + 255) & ~(size_t)255;
  float* tilepart = (float*)(ws + off);

  const int ntiles = N / TILE_M;            // 256
  const int cpc = (C + NCHUNK - 1) / NCHUNK;

  k_zero_cnt<<<(C + 255) / 256, 256, 0, stream>>>(cnt, C);
  k_hist<<<(N + 255) / 256, 256, 0, stream>>>(T, cnt, N);

  dim3 g2((D + 255) / 256, NCHUNK);
  k_wsum_part<<<g2, 256, 0, stream>>>(W, cnt, wpart, C, D, cpc);
  k_wsum_reduce<<<(D + 255) / 256, 256, 0, stream>>>(wpart, wsum, D, NCHUNK);

  k_loss_tiles<<<ntiles, 32 * WAVES_PER_TILE, 0, stream>>>(X, T, W, cnt, wsum,
                                                           tilepart, N, D);
  k_final<<<1, 256, 0, stream>>>(tilepart, ntiles, 1.0f / (float)N, out);
}